// FlashRNNTritonFused_89790586290288
// MI455X (gfx1250) — compile-verified
//
#include <hip/hip_runtime.h>
#include <math.h>

#define BB   16
#define TT   1024
#define NGG  4
#define DHH  768
#define NWG  24          // 24 WGs x 8 waves = 192 waves = 48 e-tiles x 4 gates
#define NTHR 256
#define LDSS 770         // padded stride for h_lds (even -> 8B aligned b64, spreads banks)

typedef float v2f __attribute__((ext_vector_type(2)));
typedef float v8f __attribute__((ext_vector_type(8)));

// Re-initialize workspace every launch: zero the 2048 per-step barrier counters
// and copy h0 (= states[0,0,:,:,:]) into h double-buffer slot 0.
__global__ void lstm_init_kernel(const float* __restrict__ states,
                                 float* __restrict__ h_buf,
                                 unsigned* __restrict__ bar) {
  int idx = blockIdx.x * blockDim.x + threadIdx.x;
  if (idx < 2 * TT) bar[idx] = 0u;
  if (idx < BB * DHH) h_buf[idx] = states[idx];
}

__global__ __launch_bounds__(NTHR) void lstm_persistent_kernel(
    const float* __restrict__ states,
    const float* __restrict__ Wx0, const float* __restrict__ R0, const float* __restrict__ b0,
    const float* __restrict__ Wx1, const float* __restrict__ R1, const float* __restrict__ b1,
    float* __restrict__ out, float* __restrict__ h_buf, unsigned* __restrict__ bar) {
  __shared__ float h_lds[BB][LDSS];          // current h, row-major [b][k], padded
  __shared__ float gate_lds[2][NGG][BB][16]; // pre-activation gate tiles [etile][g][b][e]
  __shared__ float c_lds[2][BB][16];         // persistent cell state for this WG's columns

  const int tid  = threadIdx.x;
  const int lane = tid & 31;
  const int wv   = tid >> 5;     // 0..7
  const int g    = wv & 3;       // gate handled by this wave
  const int wl   = wv >> 2;      // local e-tile (0..1)
  const int e0   = (blockIdx.x * 2 + wl) * 16;

  const int m    = lane & 15;          // row (A) / col (B,D) within tile
  const int koff = (lane >> 4) * 2;    // K sub-offset per ISA 16x16x4 f32 layout
  const int rowhalf = (lane >> 4) * 8; // D-matrix row half

  // init c from states[0,1,:,:,:]
  for (int p = tid; p < 2 * BB * 16; p += NTHR) {
    int w = p >> 8, rem = p & 255, bi = rem >> 4, ei = rem & 15;
    int e = (blockIdx.x * 2 + w) * 16 + ei;
    c_lds[w][bi][ei] = states[(size_t)(BB + bi) * DHH + e];
  }

  for (int layer = 0; layer < 2; ++layer) {
    const float* __restrict__ Wx = layer ? Wx1 : Wx0;
    const float* __restrict__ Rg = (layer ? R1 : R0) + (size_t)g * DHH * DHH;
    const float* __restrict__ bs = layer ? b1 : b0;

    for (int t = 0; t < TT; ++t) {
      const int s = layer * TT + t;
      const float* __restrict__ h_src = h_buf + (size_t)(s & 1) * BB * DHH;
      float* __restrict__ h_dst = h_buf + (size_t)((s & 1) ^ 1) * BB * DHH;

      // ---- Phase A: stage h[16][768] into LDS (float4 global reads) ----
      for (int p = tid; p < (BB * DHH) / 4; p += NTHR) {
        int base = p * 4;
        int bi = base / DHH, k = base % DHH;
        float4 v = *(const float4*)(h_src + base);
        h_lds[bi][k + 0] = v.x; h_lds[bi][k + 1] = v.y;
        h_lds[bi][k + 2] = v.z; h_lds[bi][k + 3] = v.w;
      }
      __syncthreads();

      // ---- Phase B: gate tile = h @ R[g][:, e0:e0+16] via fp32 WMMA 16x16x4 ----
      // A layout: lane<16 -> A[m,k..k+1]; lane>=16 -> A[m,k+2..k+3]  (koff)
      // B layout: lane<16 -> B[k..k+1, m]; lane>=16 -> B[k+2..k+3, m]
      v8f acc0 = {}, acc1 = {}, acc2 = {}, acc3 = {};
      auto mac = [&](int kk, v8f acc) -> v8f {
        const int k0 = kk * 4 + koff;
        v2f a; a.x = h_lds[m][k0]; a.y = h_lds[m][k0 + 1];
        const float* bp = Rg + (size_t)k0 * DHH + e0 + m;
        v2f bf; bf.x = bp[0]; bf.y = bp[DHH];
        return __builtin_amdgcn_wmma_f32_16x16x4_f32(false, a, false, bf,
                                                     (short)0, acc, false, false);
      };
      for (int kk = 0; kk < DHH / 4; kk += 4) {   // 192 MACs, 4 independent chains
        acc0 = mac(kk + 0, acc0);
        acc1 = mac(kk + 1, acc1);
        acc2 = mac(kk + 2, acc2);
        acc3 = mac(kk + 3, acc3);
      }
      v8f accv = (acc0 + acc1) + (acc2 + acc3);

      // ---- Phase C: share gate tiles via LDS (D layout: VGPR r -> row r+rowhalf) ----
      #pragma unroll
      for (int r = 0; r < 8; ++r)
        gate_lds[wl][g][rowhalf + r][m] = accv[r];
      __syncthreads();

      // ---- Phase D: LSTM cell update for this WG's 2 e-tiles (512 elems) ----
      for (int p = tid; p < 512; p += NTHR) {
        int w = p >> 8, rem = p & 255, bi = rem >> 4, ei = rem & 15;
        int e = (blockIdx.x * 2 + w) * 16 + ei;
        size_t wxb = (((size_t)bi * TT + t) * NGG) * DHH + e;
        float gi = gate_lds[w][0][bi][ei] + Wx[wxb + 0 * DHH] + bs[0 * DHH + e];
        float gf = gate_lds[w][1][bi][ei] + Wx[wxb + 1 * DHH] + bs[1 * DHH + e];
        float gz = gate_lds[w][2][bi][ei] + Wx[wxb + 2 * DHH] + bs[2 * DHH + e];
        float go = gate_lds[w][3][bi][ei] + Wx[wxb + 3 * DHH] + bs[3 * DHH + e];
        float ig = 1.0f / (1.0f + expf(-gi));
        float fg = 1.0f / (1.0f + expf(-gf));
        float zg = tanhf(gz);
        float og = 1.0f / (1.0f + expf(-go));
        float cn = fg * c_lds[w][bi][ei] + ig * zg;
        float hn = og * tanhf(cn);
        c_lds[w][bi][ei] = cn;
        h_dst[(size_t)bi * DHH + e] = hn;
        if (layer == 1) {
          out[(((size_t)t * 2 + 0) * BB + bi) * DHH + e] = hn;  // hs[t,0]
          out[(((size_t)t * 2 + 1) * BB + bi) * DHH + e] = cn;  // hs[t,1]
          if (t == TT - 1) {
            const size_t LH = (size_t)TT * 2 * BB * DHH;
            const size_t OO = LH + (size_t)2 * BB * DHH;
            size_t o0 = ((size_t)0 * BB + bi) * DHH + e;
            size_t o1 = ((size_t)1 * BB + bi) * DHH + e;
            out[LH + o0] = hn; out[LH + o1] = cn;   // last_h
            out[OO + o0] = hn; out[OO + o1] = cn;   // out
          }
        }
      }

      // ---- Phase E: device-wide barrier (one-shot counter per step) ----
      __threadfence();
      __syncthreads();
      if (tid == 0) {
        __hip_atomic_fetch_add(&bar[s], 1u, __ATOMIC_ACQ_REL, __HIP_MEMORY_SCOPE_AGENT);
        while (__hip_atomic_load(&bar[s], __ATOMIC_ACQUIRE, __HIP_MEMORY_SCOPE_AGENT)
               < (unsigned)NWG) {
          __builtin_amdgcn_s_sleep(1);
        }
      }
      __syncthreads();
      __threadfence();
    }
  }
}

extern "C" void kernel_launch(void* const* d_in, const int* in_sizes, int n_in,
                              void* d_out, int out_size, void* d_ws, size_t ws_size,
                              hipStream_t stream) {
  const float* states = (const float*)d_in[0];
  const float* Wx0    = (const float*)d_in[1];
  const float* R0     = (const float*)d_in[2];
  const float* b0     = (const float*)d_in[3];
  const float* Wx1    = (const float*)d_in[4];
  const float* R1     = (const float*)d_in[5];
  const float* b1     = (const float*)d_in[6];
  float* out = (float*)d_out;

  // d_ws layout: [0, 8KB) barrier counters (2048 u32), then h double buffer.
  unsigned* bar = (unsigned*)d_ws;
  float* h_buf  = (float*)((char*)d_ws + 8192);

  lstm_init_kernel<<<48, 256, 0, stream>>>(states, h_buf, bar);
  lstm_persistent_kernel<<<NWG, NTHR, 0, stream>>>(states, Wx0, R0, b0,
                                                   Wx1, R1, b1, out, h_buf, bar);
}